// QLSTM__gen099_65481071395276
// MI455X (gfx1250) — compile-verified
//
#include <hip/hip_runtime.h>

typedef __attribute__((ext_vector_type(2))) float v2f;
typedef __attribute__((ext_vector_type(8))) float v8f;

#define NT 128
#define NB 1024
#define ND 64
#define NH 8
#define WC 72          // D + H columns of each W
#define MROWS (NT*NB)  // 131072 rows of the big GEMM

// ---------------- DPP8 cross-lane helpers (groups of 8 lanes) ----------------
constexpr int dppsel(int a, int b, int c, int d, int e, int f, int g, int h) {
    return a | (b << 3) | (c << 6) | (d << 9) | (e << 12) | (f << 15) |
           (g << 18) | (h << 21);
}
constexpr int SEL_XOR1 = dppsel(1, 0, 3, 2, 5, 4, 7, 6);
constexpr int SEL_XOR2 = dppsel(2, 3, 0, 1, 6, 7, 4, 5);
constexpr int SEL_XOR4 = dppsel(4, 5, 6, 7, 0, 1, 2, 3);
constexpr int SEL_UP1  = dppsel(0, 0, 1, 2, 3, 4, 5, 6);
constexpr int SEL_UP2  = dppsel(0, 1, 0, 1, 2, 3, 4, 5);
constexpr int SEL_UP4  = dppsel(0, 1, 2, 3, 0, 1, 2, 3);

template <int SEL>
__device__ __forceinline__ float dpp8f(float x) {
    return __builtin_bit_cast(
        float, __builtin_amdgcn_mov_dpp8(__builtin_bit_cast(int, x), SEL));
}

__device__ __forceinline__ float sum8(float v) {   // all 8 lanes get the sum
    v += dpp8f<SEL_XOR1>(v);
    v += dpp8f<SEL_XOR2>(v);
    v += dpp8f<SEL_XOR4>(v);
    return v;
}

__device__ __forceinline__ float sigm(float x) {
    return 1.0f / (1.0f + __expf(-x));
}
__device__ __forceinline__ float tanh_fast(float x) {
    // 1 - 2/(e^{2x}+1): correct limits at +/-inf, no inf/inf
    float e = __expf(2.0f * x);
    return 1.0f - 2.0f / (e + 1.0f);
}

// qlayer closed form: the CNOT ring is GF(2)-linear, Z-expectations reduce to
//   out_w (w>=1) = prod_{j<=w} cos(a_j),   out_0 = prod_{j>=1} cos(a_j)
// C = cos(a_w) held by lane w of an 8-lane group.
__device__ __forceinline__ float qprod(float C, int w) {
    // inclusive prefix product over the 8-lane group (shift-up scan)
    float p = C;
    float t = dpp8f<SEL_UP1>(p); p = (w >= 1) ? p * t : p;
    t       = dpp8f<SEL_UP2>(p); p = (w >= 2) ? p * t : p;
    t       = dpp8f<SEL_UP4>(p); p = (w >= 4) ? p * t : p;
    // full product excluding wire 0 (butterfly, lane0 contributes 1.0)
    float m = (w == 0) ? 1.0f : C;
    m *= dpp8f<SEL_XOR1>(m);
    m *= dpp8f<SEL_XOR2>(m);
    m *= dpp8f<SEL_XOR4>(m);
    return (w == 0) ? m : p;
}

// ---------------------------------------------------------------------------
// Kernel 1: Xc[m][n] = sum_k x[m][k] * Wcat[n][k] + b[n] + th[n]
//   m in [0, T*B), k in [0,64), n in [0,32) (gate-major: f,i,u,o x 8 wires)
// One wave per 16-row M tile; two 16-wide N tiles; 16 K-steps of
// V_WMMA_F32_16X16X4_F32 (fp32 in / fp32 accum).
// N-tile 0 columns come from {W_f, W_i}; N-tile 1 from {W_u, W_o}:
// load both candidates from uniform global bases, select values per lane
// (keeps everything on the global path, no flat loads / pointer spills).
// ---------------------------------------------------------------------------
__global__ __launch_bounds__(256) void qlstm_xgemm(
    const float* __restrict__ x,
    const float* __restrict__ Wf, const float* __restrict__ bf, const float* __restrict__ thf,
    const float* __restrict__ Wi, const float* __restrict__ bi, const float* __restrict__ thi,
    const float* __restrict__ Wu, const float* __restrict__ bu, const float* __restrict__ thu,
    const float* __restrict__ Wo, const float* __restrict__ bo, const float* __restrict__ tho,
    float* __restrict__ xc)
{
    const int lane  = threadIdx.x & 31;
    const int wave  = blockIdx.x * (blockDim.x >> 5) + (threadIdx.x >> 5);
    const int m0    = wave << 4;
    const int lmod  = lane & 15;
    const int khalf = (lane >> 4) << 1;   // lanes 0-15 hold K={0,1}, lanes 16-31 K={2,3}
    const bool lo8  = (lmod < 8);         // first gate of the pair for this column
    const int  wrow = (lmod & 7) * WC + khalf;

    // B fragments (K x N): lane holds column n = ntile*16 + lmod, two K values.
    v2f bfr0[16], bfr1[16];
#pragma unroll
    for (int kk = 0; kk < 16; ++kk) {
        v2f vf = *(const v2f*)(Wf + wrow + kk * 4);
        v2f vi = *(const v2f*)(Wi + wrow + kk * 4);
        v2f vu = *(const v2f*)(Wu + wrow + kk * 4);
        v2f vo = *(const v2f*)(Wo + wrow + kk * 4);
        bfr0[kk] = lo8 ? vf : vi;
        bfr1[kk] = lo8 ? vu : vo;
    }
    const int wbig = lmod & 7;
    const float bias0 = lo8 ? (bf[wbig] + thf[wbig]) : (bi[wbig] + thi[wbig]);
    const float bias1 = lo8 ? (bu[wbig] + thu[wbig]) : (bo[wbig] + tho[wbig]);

    // A fragment: lane holds row m0+lmod, K = khalf + {0,1} -> aligned float2 load
    const float* xr = x + (size_t)(m0 + lmod) * ND + khalf;
    v8f c0 = {}, c1 = {};
#pragma unroll
    for (int kk = 0; kk < 16; ++kk) {
        v2f a = *(const v2f*)(xr + kk * 4);
        c0 = __builtin_amdgcn_wmma_f32_16x16x4_f32(false, a, false, bfr0[kk],
                                                   (short)0, c0, false, false);
        c1 = __builtin_amdgcn_wmma_f32_16x16x4_f32(false, a, false, bfr1[kk],
                                                   (short)0, c1, false, false);
    }

    // D layout (16x16 f32): VGPR v holds row M = v + (lane/16)*8, col N = lmod
    const int rbase = m0 + ((lane >> 4) << 3);
#pragma unroll
    for (int v = 0; v < 8; ++v) {
        xc[(size_t)(rbase + v) * 32 + lmod]      = c0[v] + bias0;
        xc[(size_t)(rbase + v) * 32 + 16 + lmod] = c1[v] + bias1;
    }
}

// ---------------------------------------------------------------------------
// Kernel 2: sequential scan over T=128, parallel over B=1024 rows.
// 8 lanes per row (one per wire); all cross-lane traffic via DPP8 (pure VALU,
// no LDS on the serial critical path). Xc (16 MB) is L2-resident.
// ---------------------------------------------------------------------------
__global__ __launch_bounds__(256) void qlstm_scan(
    const float* __restrict__ xc,
    const float* __restrict__ Wf, const float* __restrict__ Wi,
    const float* __restrict__ Wu, const float* __restrict__ Wo,
    const float* __restrict__ lnw, const float* __restrict__ lnb,
    float* __restrict__ out)
{
    const int tid = blockIdx.x * blockDim.x + threadIdx.x;   // 0..8191
    const int w   = tid & 7;      // wire handled by this lane
    const int row = tid >> 3;     // batch row

    // recurrent weights: h-columns (64..71) of this lane's wire row, 4 gates
    float whf[NH], whi[NH], whu[NH], who[NH];
#pragma unroll
    for (int j = 0; j < NH; ++j) {
        whf[j] = Wf[w * WC + ND + j];
        whi[j] = Wi[w * WC + ND + j];
        whu[j] = Wu[w * WC + ND + j];
        who[j] = Wo[w * WC + ND + j];
    }
    const float lw = lnw[w], lb = lnb[w];

    float h = 0.0f, c = 0.0f;
    for (int t = 0; t < NT; ++t) {
        const float* px = xc + (size_t)(t * NB + row) * 32;
        float af = px[w], ai = px[8 + w], au = px[16 + w], ao = px[24 + w];

        // h @ Wh.T : broadcast h_j across the 8-lane group via DPP8
#define QLSTM_HSTEP(j)                                                   \
        { float hj = dpp8f<(0x249249 * j)>(h);                           \
          af = fmaf(hj, whf[j], af); ai = fmaf(hj, whi[j], ai);          \
          au = fmaf(hj, whu[j], au); ao = fmaf(hj, who[j], ao); }
        QLSTM_HSTEP(0) QLSTM_HSTEP(1) QLSTM_HSTEP(2) QLSTM_HSTEP(3)
        QLSTM_HSTEP(4) QLSTM_HSTEP(5) QLSTM_HSTEP(6) QLSTM_HSTEP(7)
#undef QLSTM_HSTEP

        float gf = sigm(qprod(__cosf(af), w));
        float gi = sigm(qprod(__cosf(ai), w));
        float gu = tanh_fast(qprod(__cosf(au), w));
        float go = sigm(qprod(__cosf(ao), w));
        c = gf * c + gi * gu;
        float hr = go * tanh_fast(c);

        // layernorm across the 8 wires of this row (DPP8 butterfly sums)
        float mu = sum8(hr) * 0.125f;
        float d  = hr - mu;
        float var = sum8(d * d) * 0.125f;
        float rs  = rsqrtf(var + 1e-5f);
        h = d * rs * lw + lb;

        out[(size_t)(t * NB + row) * NH + w] = h;
    }
    // hx, cx appended after outputs
    out[(size_t)NT * NB * NH + (size_t)row * NH + w] = h;
    out[(size_t)NT * NB * NH + (size_t)NB * NH + (size_t)row * NH + w] = c;
}

extern "C" void kernel_launch(void* const* d_in, const int* in_sizes, int n_in,
                              void* d_out, int out_size, void* d_ws, size_t ws_size,
                              hipStream_t stream) {
    const float* x   = (const float*)d_in[0];
    const float* Wf  = (const float*)d_in[1];
    const float* bf  = (const float*)d_in[2];
    const float* thf = (const float*)d_in[3];
    const float* Wi  = (const float*)d_in[4];
    const float* bi  = (const float*)d_in[5];
    const float* thi = (const float*)d_in[6];
    const float* Wu  = (const float*)d_in[7];
    const float* bu  = (const float*)d_in[8];
    const float* thu = (const float*)d_in[9];
    const float* Wo  = (const float*)d_in[10];
    const float* bo  = (const float*)d_in[11];
    const float* tho = (const float*)d_in[12];
    const float* lnw = (const float*)d_in[13];
    const float* lnb = (const float*)d_in[14];
    float* out = (float*)d_out;
    float* xcw = (float*)d_ws;   // 131072 x 32 floats = 16 MB scratch

    // 8192 M-tiles of 16 rows, 8 waves (256 threads) per block -> 1024 blocks
    qlstm_xgemm<<<dim3(MROWS / 16 / 8), dim3(256), 0, stream>>>(
        x, Wf, bf, thf, Wi, bi, thi, Wu, bu, thu, Wo, bo, tho, xcw);

    // 1024 rows x 8 lanes = 8192 threads -> 32 blocks of 256
    qlstm_scan<<<dim3((NB * NH) / 256), dim3(256), 0, stream>>>(
        xcw, Wf, Wi, Wu, Wo, lnw, lnb, out);
}